// TrajLoss_33097017983511
// MI455X (gfx1250) — compile-verified
//
#include <hip/hip_runtime.h>

// ---------------------------------------------------------------------------
// TrajLoss: pos = cumsum(interleaved (dx,dy) pairs); sum((pos-x_true)^2 + ...)
// B=8192 rows, 1024 pairs/row. Memory-bound: 128 MiB read -> ~5.8us floor
// at 23.3 TB/s. One wave32 per row; CDNA5 async global->LDS double buffering
// (GLOBAL_LOAD_ASYNC_TO_LDS_B128 + s_wait_asynccnt) for the streamed operand.
// traj_du_true (d_in[1]) is unused by the reference and is never read.
// ---------------------------------------------------------------------------

#define ROWS_PER_BLOCK 8
#define ROW_FLOATS     2048
#define TPAIRS         1024
#define CHUNKS         8      // 1024 pairs / 128 pairs per chunk
#define CHUNK_FLOATS   256    // 128 pairs = 1 KiB

// Issue one chunk (1 KiB for this wave): each lane DMAs its own 32 bytes of
// predictions into its LDS slot via two async b128 ops. INST_OFFSET is added
// to BOTH the global and LDS addresses (ISA 15.18.3), so offset:16 covers the
// second half of the lane's 32B. s_wait_dscnt 0x0 retires the previous
// iteration's ds reads of the buffer we are about to overwrite (WAR guard).
__device__ __forceinline__ void async_pred_chunk(const float* prow,
                                                 const float* lds_slot,
                                                 unsigned global_byte_off) {
  unsigned lds_off = (unsigned)(size_t)(const void*)lds_slot; // lo32 = LDS offset
  asm volatile(
      "s_wait_dscnt 0x0\n"
      "global_load_async_to_lds_b128 %0, %1, %2\n"
      "global_load_async_to_lds_b128 %0, %1, %2 offset:16"
      :
      : "v"(lds_off), "v"(global_byte_off), "s"(prow)
      : "memory");
}

extern "C" __global__ void __launch_bounds__(256)
traj_loss_stage1(const float* __restrict__ pred,
                 const float* __restrict__ tpos,
                 float* __restrict__ partial) {
  __shared__ __align__(16) float smem[ROWS_PER_BLOCK * 2 * CHUNK_FLOATS]; // 16 KiB
  __shared__ float wavesum[ROWS_PER_BLOCK];

  const int  lane = threadIdx.x & 31;
  const int  wave = threadIdx.x >> 5;
  const long row  = (long)blockIdx.x * ROWS_PER_BLOCK + wave;

  const float* prow = pred + row * ROW_FLOATS;
  const float* trow = tpos + row * ROW_FLOATS;

  float* mybuf   = &smem[wave * (2 * CHUNK_FLOATS)];
  float* slot0   = mybuf + lane * 8;                 // lane's 8 floats, buffer 0
  float* slot1   = mybuf + CHUNK_FLOATS + lane * 8;  // lane's 8 floats, buffer 1

  float carry_x = 0.f, carry_y = 0.f, acc = 0.f;

  // Prologue: chunk 0 -> buffer 0
  async_pred_chunk(prow, slot0, (unsigned)(lane * 32));

#pragma unroll
  for (int c = 0; c < CHUNKS; ++c) {
    const float* slot = (c & 1) ? slot1 : slot0;

    // Prefetch chunk c+1 into the other buffer (hidden behind this chunk's math).
    if (c + 1 < CHUNKS) {
      const float* nslot = ((c + 1) & 1) ? slot1 : slot0;
      async_pred_chunk(prow, nslot, (unsigned)((c + 1) * 1024 + lane * 32));
    }

    // True positions for my 4 pairs (t = 128c + 4*lane + k). Issued before the
    // async wait so the normal loads overlap the DMA wait.
    const int t0 = c * 128 + lane * 4;
    float4 tx = *(const float4*)(trow + t0);
    float4 ty = *(const float4*)(trow + TPAIRS + t0);

    if (c + 1 < CHUNKS)
      asm volatile("s_wait_asynccnt 0x2" ::: "memory"); // current chunk landed
    else
      asm volatile("s_wait_asynccnt 0x0" ::: "memory"); // last chunk: drain

    // My 4 (dx,dy) pairs from LDS.
    float4 a = *(const float4*)(slot);
    float4 b = *(const float4*)(slot + 4);

    // Within-lane inclusive prefix over 4 pairs.
    float cx0 = a.x,        cy0 = a.y;
    float cx1 = cx0 + a.z,  cy1 = cy0 + a.w;
    float cx2 = cx1 + b.x,  cy2 = cy1 + b.y;
    float cx3 = cx2 + b.z,  cy3 = cy2 + b.w;

    // Wave32 inclusive scan of the lane totals.
    float sx = cx3, sy = cy3;
#pragma unroll
    for (int off = 1; off < 32; off <<= 1) {
      float ux = __shfl_up(sx, off, 32);
      float uy = __shfl_up(sy, off, 32);
      if (lane >= off) { sx += ux; sy += uy; }
    }

    // Exclusive offset for this lane + running carry from previous chunks.
    float bx = carry_x + (sx - cx3);
    float by = carry_y + (sy - cy3);
    carry_x += __shfl(sx, 31, 32);   // chunk total (uses OLD carry above)
    carry_y += __shfl(sy, 31, 32);

    float ex, ey;
    ex = bx + cx0 - tx.x; ey = by + cy0 - ty.x; acc = fmaf(ex, ex, fmaf(ey, ey, acc));
    ex = bx + cx1 - tx.y; ey = by + cy1 - ty.y; acc = fmaf(ex, ex, fmaf(ey, ey, acc));
    ex = bx + cx2 - tx.z; ey = by + cy2 - ty.z; acc = fmaf(ex, ex, fmaf(ey, ey, acc));
    ex = bx + cx3 - tx.w; ey = by + cy3 - ty.w; acc = fmaf(ex, ex, fmaf(ey, ey, acc));
  }

  // Wave reduction, then block combine (atomic-free, deterministic).
#pragma unroll
  for (int off = 16; off; off >>= 1) acc += __shfl_xor(acc, off, 32);
  if (lane == 0) wavesum[wave] = acc;
  __syncthreads();
  if (threadIdx.x == 0) {
    float s = 0.f;
#pragma unroll
    for (int w = 0; w < ROWS_PER_BLOCK; ++w) s += wavesum[w];
    partial[blockIdx.x] = s;
  }
}

extern "C" __global__ void __launch_bounds__(256)
traj_loss_stage2(const float* __restrict__ partial, float* __restrict__ out, int n) {
  __shared__ double sm[256];
  double s = 0.0;
  for (int i = threadIdx.x; i < n; i += 256) s += (double)partial[i];
  sm[threadIdx.x] = s;
  __syncthreads();
  for (int stride = 128; stride; stride >>= 1) {
    if ((int)threadIdx.x < stride) sm[threadIdx.x] += sm[threadIdx.x + stride];
    __syncthreads();
  }
  if (threadIdx.x == 0) out[0] = (float)sm[0];
}

extern "C" void kernel_launch(void* const* d_in, const int* in_sizes, int n_in,
                              void* d_out, int out_size, void* d_ws, size_t ws_size,
                              hipStream_t stream) {
  (void)n_in; (void)out_size; (void)ws_size;
  const float* pred = (const float*)d_in[0];  // traj_pred
  // d_in[1] = traj_du_true: unused by the reference -> never touched.
  const float* tpos = (const float*)d_in[2];  // traj_pos_true
  float* out     = (float*)d_out;
  float* partial = (float*)d_ws;              // 1024 floats of scratch

  const int rows   = in_sizes[0] / ROW_FLOATS;     // 8192
  const int blocks = rows / ROWS_PER_BLOCK;        // 1024

  traj_loss_stage1<<<blocks, 256, 0, stream>>>(pred, tpos, partial);
  traj_loss_stage2<<<1, 256, 0, stream>>>(partial, out, blocks);
}